// EdgeSHLayer_4114578670342
// MI455X (gfx1250) — compile-verified
//
#include <hip/hip_runtime.h>
#include <cstdint>
#include <cstddef>

#define THREADS 256
#define NBASIS 10
#define NSH 9
#define PFDIST 16384

// _SMOOTH_C * sqrt(NUM_BASIS) = 1.14136 * e^2 * sqrt(10)
#define KSCALE ((float)(1.14136 * 7.38905609893065 * 3.1622776601683795))
#define SQRT3F 1.7320508075688772f

__device__ __forceinline__ void compute_edge(
    const float* __restrict__ pos,
    const int*   __restrict__ srcp,
    const int*   __restrict__ dstp,
    float invstep, int i,
    float emb[NBASIS], float sh[NSH])
{
    // Streamed once -> non-temporal loads keep pos hot in L2.
    const int s = __builtin_nontemporal_load(srcp + i);
    const int d = __builtin_nontemporal_load(dstp + i);

    // pos fits in L2 (1.2 MB of 192 MB) -> gathers are cheap after warmup.
    const float ax = pos[3*s+0], ay = pos[3*s+1], az = pos[3*s+2];
    const float bx = pos[3*d+0], by = pos[3*d+1], bz = pos[3*d+2];

    const float vx = bx - ax, vy = by - ay, vz = bz - az;
    const float r  = sqrtf(vx*vx + vy*vy + vz*vz);
    const float inv = __builtin_amdgcn_rcpf(fmaxf(r, 1e-12f));
    const float x = vx * inv, y = vy * inv, z = vz * inv;

    sh[0] = 1.0f;
    sh[1] = x; sh[2] = y; sh[3] = z;
    sh[4] = SQRT3F * x * z;
    sh[5] = SQRT3F * x * y;
    sh[6] = y*y - 0.5f * (x*x + z*z);
    sh[7] = SQRT3F * y * z;
    sh[8] = 0.5f * SQRT3F * (z*z - x*x);

    // smooth_finite basis: step = end/11, d_k = r/step - k (k = 1..10)
    // sus(1+d)*sus(1-d) == exp(-2/(1-d^2)) for |d|<1, else 0
    const float t = r * invstep;
    #pragma unroll
    for (int k = 0; k < NBASIS; ++k) {
        const float dk = t - (float)(k + 1);
        const float g  = 1.0f - dk * dk;
        float v = 0.0f;
        if (g > 0.0f)
            v = KSCALE * __expf(-2.0f * __builtin_amdgcn_rcpf(g));
        emb[k] = v;
    }
}

__global__ __launch_bounds__(THREADS) void edge_sh_kernel(
    const float* __restrict__ pos,      // [N,3]
    const float* __restrict__ radius,   // scalar
    const int*   __restrict__ edge,     // [2,E]
    float*       __restrict__ out,      // emb [E,10] then sh [E,9]
    int E)
{
    __shared__ __align__(16) float s_emb[THREADS * NBASIS]; // 10240 B
    __shared__ __align__(16) float s_sh [THREADS * NSH];    //  9216 B

    const int e0  = (int)blockIdx.x * THREADS;
    const int tid = (int)threadIdx.x;
    const int i   = e0 + tid;

    const int* __restrict__ srcp = edge;
    const int* __restrict__ dstp = edge + E;

    const float end     = radius[0];              // uniform -> s_load
    const float invstep = (float)(NBASIS + 1) / end;

    float emb[NBASIS];
    float sh[NSH];

    if (e0 + THREADS <= E) {
        // ---- full block: unpredicated compute, LDS-staged async DMA stores ----
        if (e0 + THREADS + PFDIST <= E) {         // uniform branch, no per-lane clamp
            __builtin_prefetch(srcp + i + PFDIST, 0, 0);
            __builtin_prefetch(dstp + i + PFDIST, 0, 0);
        }

        compute_edge(pos, srcp, dstp, invstep, i, emb, sh);

        // Stage rows in LDS (strides 10 and 9 dwords vs 64 banks: conflict-free).
        #pragma unroll
        for (int k = 0; k < NBASIS; ++k) s_emb[tid * NBASIS + k] = emb[k];
        #pragma unroll
        for (int k = 0; k < NSH; ++k)    s_sh [tid * NSH   + k] = sh[k];
        __syncthreads();

        float* out_emb = out;
        float* out_sh  = out + (size_t)E * NBASIS;

        const unsigned embBytes = (unsigned)e0 * (NBASIS * 4u); // 16B aligned
        const unsigned shBytes  = (unsigned)e0 * (NSH * 4u);    // 16B aligned
        const unsigned sEmbBase = (unsigned)(uintptr_t)&s_emb[0];
        const unsigned sShBase  = (unsigned)(uintptr_t)&s_sh[0];

        // Fully-coalesced 16B/lane async DMA stores LDS -> global, write-once
        // stream marked non-temporal. emb: 640 x b128, sh: 576 x b128 per block.
        for (int q = tid; q < (THREADS * NBASIS) / 4; q += THREADS) {
            const unsigned goff = embBytes + 16u * (unsigned)q;
            const unsigned loff = sEmbBase + 16u * (unsigned)q;
            asm volatile("global_store_async_from_lds_b128 %0, %1, %2 th:TH_STORE_NT"
                         :: "v"(goff), "v"(loff), "s"(out_emb)
                         : "memory");
        }
        for (int q = tid; q < (THREADS * NSH) / 4; q += THREADS) {
            const unsigned goff = shBytes + 16u * (unsigned)q;
            const unsigned loff = sShBase + 16u * (unsigned)q;
            asm volatile("global_store_async_from_lds_b128 %0, %1, %2 th:TH_STORE_NT"
                         :: "v"(goff), "v"(loff), "s"(out_sh)
                         : "memory");
        }
        // DMA engine rereads LDS; drain before the block's LDS is deallocated.
        asm volatile("s_wait_asynccnt 0" ::: "memory");
    } else if (i < E) {
        // ---- tail block: guarded compute, direct scalar stores ----
        compute_edge(pos, srcp, dstp, invstep, i, emb, sh);
        float* out_emb = out + (size_t)i * NBASIS;
        #pragma unroll
        for (int k = 0; k < NBASIS; ++k) out_emb[k] = emb[k];
        float* out_sh = out + (size_t)E * NBASIS + (size_t)i * NSH;
        #pragma unroll
        for (int k = 0; k < NSH; ++k) out_sh[k] = sh[k];
    }
}

extern "C" void kernel_launch(void* const* d_in, const int* in_sizes, int n_in,
                              void* d_out, int out_size, void* d_ws, size_t ws_size,
                              hipStream_t stream) {
    // setup_inputs order: feature (unused), pos, max_neighbor_radius, edge
    const float* pos    = (const float*)d_in[1];
    const float* radius = (const float*)d_in[2];
    const int*   edge   = (const int*)d_in[3];
    const int    E      = in_sizes[3] / 2;

    float* out = (float*)d_out;
    const int blocks = (E + THREADS - 1) / THREADS;
    edge_sh_kernel<<<blocks, THREADS, 0, stream>>>(pos, radius, edge, out, E);
}